// WeightedPartialAttention_35407710388233
// MI455X (gfx1250) — compile-verified
//
#include <hip/hip_runtime.h>

// Problem constants (match reference setup_inputs)
#define B_DIM  32
#define C_DIM  256
#define H_DIM  56
#define W_DIM  56
#define HW_DIM (H_DIM * W_DIM)       // 3136
#define K_TOP  (HW_DIM / 2)          // 1568  (MASKING_RATIO = 0.5)
#define ALPHA_C 0.6f
#define BETA_C  0.2f
#define GAMMA_C 0.2f

typedef float v2f __attribute__((ext_vector_type(2)));
typedef float v8f __attribute__((ext_vector_type(8)));

__device__ __forceinline__ unsigned f2key(float f) {
  // Monotonic float -> uint mapping (larger float => larger key)
  unsigned u = __float_as_uint(f);
  return (u & 0x80000000u) ? ~u : (u | 0x80000000u);
}

__device__ __forceinline__ float sigmoidf_fast(float v) {
  return 1.0f / (1.0f + __expf(-v));
}

// ---------------------------------------------------------------------------
// Pass 1: combined score per (b, hw).
// Channel reduction sum_c x^2 done with V_WMMA_F32_16X16X4_F32:
//   A (16x4 f32) = x^2 tile (16 pixels x 4 channels), B (4x16) = ones
//   => each row of D accumulates the row-sum of A in exact fp32.
// Block = 128 threads = 4 waves; each wave owns 16 pixels; block owns 64.
// Grid = (HW/64, B). HW = 3136 = 64*49 exactly -> EXEC all-1s everywhere.
// ---------------------------------------------------------------------------
__global__ void __launch_bounds__(128)
score_kernel(const float* __restrict__ x,
             const float* __restrict__ gaze,
             const float* __restrict__ pose,
             float* __restrict__ scores) {
  const int b    = blockIdx.y;
  const int wave = threadIdx.x >> 5;
  const int lane = threadIdx.x & 31;
  const int half = lane >> 4;                    // 0: K={0,1}, 1: K={2,3}
  const int p_base = blockIdx.x * 64 + wave * 16;
  const int pix    = p_base + (lane & 15);       // M = lane % 16

  const float* __restrict__ xb = x + (size_t)b * C_DIM * HW_DIM;

  v2f ones; ones.x = 1.0f; ones.y = 1.0f;
  v8f acc = {0.f, 0.f, 0.f, 0.f, 0.f, 0.f, 0.f, 0.f};

  for (int c = 0; c < C_DIM; c += 4) {
    const int c0 = c + half * 2;
    const float v0 = xb[(size_t)c0 * HW_DIM + pix];        // K = c0   - c
    const float v1 = xb[(size_t)(c0 + 1) * HW_DIM + pix];  // K = c0+1 - c
    v2f a; a.x = v0 * v0; a.y = v1 * v1;
    // emits v_wmma_f32_16x16x4_f32 (accumulating into acc)
    acc = __builtin_amdgcn_wmma_f32_16x16x4_f32(
        /*neg_a=*/false, a, /*neg_b=*/false, ones,
        /*c_mod=*/(short)0, acc, /*reuse_a=*/false, /*reuse_b=*/false);
  }

  // D layout: acc[r] = pixel p_base+r (lanes 0-15), p_base+8+r (lanes 16-31),
  // replicated across N. Lanes 0-7 and 16-23 each emit one score.
  const int r = lane & 7;
  float s = acc[0];
  s = (r == 1) ? acc[1] : s;
  s = (r == 2) ? acc[2] : s;
  s = (r == 3) ? acc[3] : s;
  s = (r == 4) ? acc[4] : s;
  s = (r == 5) ? acc[5] : s;
  s = (r == 6) ? acc[6] : s;
  s = (r == 7) ? acc[7] : s;

  const bool active = (lane < 8) || (lane >= 16 && lane < 24);
  if (active) {
    const int q = p_base + ((lane >= 16) ? 8 : 0) + r;
    const size_t o = (size_t)b * HW_DIM + q;
    const float energy = __builtin_sqrtf(s);
    const float sc = ALPHA_C * energy
                   + BETA_C  * sigmoidf_fast(gaze[o])
                   + GAMMA_C * sigmoidf_fast(pose[o]);
    scores[o] = sc;
  }
}

// ---------------------------------------------------------------------------
// Pass 2: exact per-batch top-K threshold via 4-round MSB radix select on
// monotone uint keys, then write the 0/1 mask. One 256-thread block per batch.
// ---------------------------------------------------------------------------
__global__ void __launch_bounds__(256)
select_mask_kernel(const float* __restrict__ scores,
                   float* __restrict__ mask) {
  const int b   = blockIdx.x;
  const int tid = threadIdx.x;
  const float* __restrict__ row = scores + (size_t)b * HW_DIM;

  __shared__ unsigned hist[256];
  __shared__ unsigned prefix_s;
  __shared__ int kk_s;
  __shared__ int tiectr;

  if (tid == 0) { prefix_s = 0u; kk_s = K_TOP; tiectr = 0; }
  __syncthreads();

  for (int pass = 0; pass < 4; ++pass) {
    const int shift = 24 - 8 * pass;
    const unsigned himask = (pass == 0) ? 0u : (0xFFFFFFFFu << (shift + 8));
    const unsigned prefix = prefix_s;

    hist[tid] = 0u;
    __syncthreads();

    for (int i = tid; i < HW_DIM; i += 256) {
      const unsigned key = f2key(row[i]);
      if ((key & himask) == prefix) {
        atomicAdd(&hist[(key >> shift) & 0xFFu], 1u);
      }
    }
    __syncthreads();

    if (tid == 0) {
      unsigned cum = 0;
      int bsel = 0;
      const int kk = kk_s;
      for (int v = 255; v >= 0; --v) {
        if ((int)(cum + hist[v]) >= kk) { bsel = v; break; }
        cum += hist[v];
      }
      prefix_s = prefix | ((unsigned)bsel << shift);
      kk_s = kk - (int)cum;     // rank of threshold within its tie group
    }
    __syncthreads();
  }

  const unsigned T = prefix_s;   // exact K-th largest key
  const int ties_to_take = kk_s; // how many key==T elements belong in top-K
  float* __restrict__ mrow = mask + (size_t)b * HW_DIM;

  for (int i = tid; i < HW_DIM; i += 256) {
    const unsigned key = f2key(row[i]);
    float m;
    if (key > T) {
      m = 1.0f;
    } else if (key == T) {
      m = (atomicAdd(&tiectr, 1) < ties_to_take) ? 1.0f : 0.0f;
    } else {
      m = 0.0f;
    }
    mrow[i] = m;
  }
}

// ---------------------------------------------------------------------------
// Pass 3: out = x * mask (broadcast over channels), float4-vectorized.
// HW = 3136 = 784 float4s per (b,c) row.
// ---------------------------------------------------------------------------
__global__ void __launch_bounds__(256)
gate_kernel(const float4* __restrict__ x4,
            const float4* __restrict__ mask4,
            float4* __restrict__ out4,
            int total4) {
  const int idx = blockIdx.x * blockDim.x + threadIdx.x;
  if (idx >= total4) return;
  const int hw4pb = HW_DIM / 4;                 // 784
  const int bc  = idx / hw4pb;
  const int hw4 = idx - bc * hw4pb;
  const int b   = bc / C_DIM;

  const float4 xv = x4[idx];
  const float4 mv = mask4[(size_t)b * hw4pb + hw4];
  float4 o;
  o.x = xv.x * mv.x;
  o.y = xv.y * mv.y;
  o.z = xv.z * mv.z;
  o.w = xv.w * mv.w;
  out4[idx] = o;
}

// ---------------------------------------------------------------------------
extern "C" void kernel_launch(void* const* d_in, const int* in_sizes, int n_in,
                              void* d_out, int out_size, void* d_ws, size_t ws_size,
                              hipStream_t stream) {
  const float* x    = (const float*)d_in[0];
  const float* gaze = (const float*)d_in[1];
  const float* pose = (const float*)d_in[2];
  float* out = (float*)d_out;

  // Workspace layout: [scores B*HW floats][mask B*HW floats]
  float* scores = (float*)d_ws;
  float* maskp  = scores + (size_t)B_DIM * HW_DIM;

  // Pass 1: scores
  {
    dim3 grid(HW_DIM / 64, B_DIM);
    score_kernel<<<grid, 128, 0, stream>>>(x, gaze, pose, scores);
  }
  // Pass 2: exact top-K mask per batch
  {
    select_mask_kernel<<<B_DIM, 256, 0, stream>>>(scores, maskp);
  }
  // Pass 3: gate
  {
    const int total4 = B_DIM * C_DIM * (HW_DIM / 4);
    const int blocks = (total4 + 255) / 256;
    gate_kernel<<<blocks, 256, 0, stream>>>((const float4*)x, (const float4*)maskp,
                                            (float4*)out, total4);
  }
}